// SurfaceConv_72825465471205
// MI455X (gfx1250) — compile-verified
//
#include <hip/hip_runtime.h>

typedef float v2f __attribute__((ext_vector_type(2)));
typedef float v4f __attribute__((ext_vector_type(4)));
typedef float v8f __attribute__((ext_vector_type(8)));

#define NPTS   16384            // N (points per batch)
#define BB     4                // batch
#define CC     64               // channels
#define KK     16               // neighbors
#define OO     16               // weightnet out
#define C2     1024             // OO*CC (linear K-dim)
#define NP_TOT (BB*NPTS)        // 65536 total points
#define PTS_WG 128              // points per workgroup
#define THREADS 128             // 4 waves (wave32)
#define PANELS  (PTS_WG/16)     // 8 WMMA B-panels
#define XYZ_N  (BB*NPTS*3)      // 196608 floats of xyz passthrough

#define WSTR   260              // Wbuf row stride (floats): 16B-aligned rows, bank offset 4/lane
#define VSTR   (2*PTS_WG + 32)  // Vbuf row stride (floats): half-wave windows disjoint mod 64 banks

// ---------------------------------------------------------------------------
// Fused: weightnet (VALU) -> per-point 64x16x16 agg (VALU, point-per-lane)
//        -> 64x1024 GEMM over points (f32 WMMA) -> ws[channel][point] (+lin_b)
// ---------------------------------------------------------------------------
__global__ __launch_bounds__(THREADS)
void surfconv_fused(const float* __restrict__ points,
                    const float* __restrict__ coordinate,
                    const float* __restrict__ w1,
                    const float* __restrict__ b1,
                    const float* __restrict__ lin_w,
                    const float* __restrict__ lin_b,
                    float* __restrict__ out2)           // [CC][NP_TOT]
{
    __shared__ float Wbuf[PTS_WG * WSTR];               // [pt][o*16+k]   130 KB, b128 access
    __shared__ float Vbuf[2][8][VSTR];                  // [opair][pt*2+(o&1)]  2 x 9 KB

    const int tid   = threadIdx.x;
    const int nbase = blockIdx.x * PTS_WG;              // global point base (b uniform: 128 | 16384)
    const int b     = nbase >> 14;                      // / NPTS
    const int n     = (nbase & (NPTS - 1)) + tid;       // point index within batch
    const int pt    = tid;

    const float* __restrict__ coordB = coordinate + (size_t)b * 3  * KK * NPTS;
    const float* __restrict__ ptsB   = points     + (size_t)b * CC * KK * NPTS;

    // ---- stage 1: W[o][k] = relu(w1[o,:] . coord[:,k] + b1[o]) -> LDS ------
    float cv[3][KK];
#pragma unroll
    for (int c3 = 0; c3 < 3; ++c3)
#pragma unroll
        for (int k = 0; k < KK; ++k)
            cv[c3][k] = coordB[(size_t)(c3 * KK + k) * NPTS + n];

#pragma unroll
    for (int o = 0; o < OO; ++o) {
        const float wa = w1[o * 3 + 0], wb = w1[o * 3 + 1], wc = w1[o * 3 + 2];
        const float bo = b1[o];
#pragma unroll
        for (int k4 = 0; k4 < 4; ++k4) {
            v4f wv;
#pragma unroll
            for (int j = 0; j < 4; ++j) {
                const int k = k4 * 4 + j;
                float w = fmaf(wa, cv[0][k], fmaf(wb, cv[1][k], fmaf(wc, cv[2][k], bo)));
                wv[j] = fmaxf(w, 0.0f);
            }
            *(v4f*)&Wbuf[pt * WSTR + o * KK + k4 * 4] = wv;   // ds_store_b128
        }
    }
    __syncthreads();

    // ---- stage 3 wave roles -----------------------------------------------
    const int lane = tid & 31;
    const int m    = tid >> 5;          // wave id == M-tile of lin_w (rows m*16..m*16+15)
    const int half = lane >> 4;         // 0: lanes 0-15, 1: lanes 16-31
    const int l16  = lane & 15;

    v8f acc[PANELS];
#pragma unroll
    for (int p = 0; p < PANELS; ++p)
        acc[p] = v8f{0.f, 0.f, 0.f, 0.f, 0.f, 0.f, 0.f, 0.f};

    for (int c = 0; c < CC; ++c) {
        const int buf = c & 1;

        // -- stage 2: V[o] = sum_k P[c,k] * W[o,k]  (per-lane point) --------
        float pr[KK];
#pragma unroll
        for (int k = 0; k < KK; ++k)
            pr[k] = ptsB[(size_t)(c * KK + k) * NPTS + n];
        if (c + 1 < CC)
            __builtin_prefetch(&ptsB[(size_t)((c + 1) * KK) * NPTS + n], 0, 1);

#pragma unroll
        for (int op = 0; op < 8; ++op) {                 // o-pair: o = 2*op, 2*op+1
            v2f vv = {0.f, 0.f};
#pragma unroll
            for (int k4 = 0; k4 < 4; ++k4) {
                const v4f w0 = *(const v4f*)&Wbuf[pt * WSTR + (2 * op    ) * KK + k4 * 4];
                const v4f w1v= *(const v4f*)&Wbuf[pt * WSTR + (2 * op + 1) * KK + k4 * 4];
#pragma unroll
                for (int j = 0; j < 4; ++j) {
                    const float p4 = pr[k4 * 4 + j];
                    vv.x = fmaf(p4, w0[j],  vv.x);
                    vv.y = fmaf(p4, w1v[j], vv.y);
                }
            }
            *(v2f*)&Vbuf[buf][op][pt * 2] = vv;          // ds_store_b64 (interleaved by o-parity)
        }
        __syncthreads();   // single barrier/iter is safe with double buffer

        // -- stage 3: acc[p] += lin_w[m-tile, c*16+4t..] @ Vchunk -----------
#pragma unroll
        for (int t = 0; t < 4; ++t) {
            const int kb  = c * 16 + 4 * t;              // k' base
            const int row = m * 16 + l16;                // A: lanes = M rows
            // A pair: VGPR0 = K(kb+2h), VGPR1 = K(kb+2h+1)  -> one global b64
            const v2f a = *(const v2f*)&lin_w[row * C2 + kb + 2 * half];
            const int rp = 2 * t + half;                 // o-pair row = (4t+2h)/2
#pragma unroll
            for (int p = 0; p < PANELS; ++p) {
                // B pair contiguous in LDS -> one ds_load_b64
                const v2f bv = *(const v2f*)&Vbuf[buf][rp][(p * 16 + l16) * 2];
                acc[p] = __builtin_amdgcn_wmma_f32_16x16x4_f32(
                    false, a, false, bv, (short)0, acc[p], false, false);
            }
        }
    }

    // ---- store: C/D layout lane=N(point), VGPR r -> M = r + 8*half --------
#pragma unroll
    for (int p = 0; p < PANELS; ++p) {
#pragma unroll
        for (int r = 0; r < 8; ++r) {
            const int o2  = m * 16 + half * 8 + r;
            const int col = nbase + p * 16 + l16;
            out2[(size_t)o2 * NP_TOT + col] = acc[p][r] + lin_b[o2];
        }
    }
}

// ---------------------------------------------------------------------------
// BN reduction: one block per channel -> sum, sumsq (deterministic, b128 loads)
// ---------------------------------------------------------------------------
__global__ __launch_bounds__(256)
void bn_reduce(const float* __restrict__ out2, float* __restrict__ sums)
{
    const int c = blockIdx.x;
    const v4f* __restrict__ row = (const v4f*)(out2 + (size_t)c * NP_TOT);
    float s = 0.f, s2 = 0.f;
    for (int i = threadIdx.x; i < NP_TOT / 4; i += 256) {
        const v4f x = row[i];
#pragma unroll
        for (int j = 0; j < 4; ++j) {
            s += x[j];
            s2 = fmaf(x[j], x[j], s2);
        }
    }
    __shared__ float ls[256], ls2[256];
    ls[threadIdx.x]  = s;
    ls2[threadIdx.x] = s2;
    __syncthreads();
    for (int off = 128; off > 0; off >>= 1) {
        if (threadIdx.x < off) {
            ls[threadIdx.x]  += ls[threadIdx.x + off];
            ls2[threadIdx.x] += ls2[threadIdx.x + off];
        }
        __syncthreads();
    }
    if (threadIdx.x == 0) {
        sums[c]      = ls[0];
        sums[CC + c] = ls2[0];
    }
}

// ---------------------------------------------------------------------------
// BN normalize + ReLU + write [B,C,N] output (b128 in/out)
// ---------------------------------------------------------------------------
__global__ __launch_bounds__(256)
void bn_normalize(const float* __restrict__ out2, const float* __restrict__ sums,
                  const float* __restrict__ gamma, const float* __restrict__ beta,
                  float* __restrict__ out)
{
    const int idx4 = blockIdx.x * 256 + threadIdx.x;    // over CC*NP_TOT/4
    if (idx4 >= CC * NP_TOT / 4) return;
    const int idx = idx4 * 4;
    const int c   = idx >> 16;                          // / NP_TOT (65536)
    const int np  = idx & (NP_TOT - 1);
    const float inv   = 1.0f / (float)NP_TOT;
    const float mean  = sums[c] * inv;
    const float var   = fmaf(-mean, mean, sums[CC + c] * inv);
    const float scale = rsqrtf(var + 1e-5f) * gamma[c];
    const float bias  = beta[c];
    const v4f x = *(const v4f*)(out2 + idx);
    v4f y;
#pragma unroll
    for (int j = 0; j < 4; ++j)
        y[j] = fmaxf(fmaf(x[j] - mean, scale, bias), 0.0f);
    const int b = np >> 14;                             // / NPTS
    const int n = np & (NPTS - 1);
    *(v4f*)(out + (size_t)(b * CC + c) * NPTS + n) = y;
}

// ---------------------------------------------------------------------------
extern "C" void kernel_launch(void* const* d_in, const int* in_sizes, int n_in,
                              void* d_out, int out_size, void* d_ws, size_t ws_size,
                              hipStream_t stream)
{
    const float* xyz        = (const float*)d_in[0];
    const float* points     = (const float*)d_in[1];
    const float* coordinate = (const float*)d_in[2];
    const float* w1         = (const float*)d_in[3];
    const float* b1         = (const float*)d_in[4];
    const float* lin_w      = (const float*)d_in[5];
    const float* lin_b      = (const float*)d_in[6];
    const float* gamma      = (const float*)d_in[7];
    const float* beta       = (const float*)d_in[8];

    float* out  = (float*)d_out;
    float* out2 = (float*)d_ws;                         // CC*NP_TOT floats (16 MB)
    float* sums = out2 + (size_t)CC * NP_TOT;           // 2*CC floats

    // output tuple: (xyz, new_points) flat-concatenated
    hipMemcpyAsync(out, xyz, (size_t)XYZ_N * sizeof(float),
                   hipMemcpyDeviceToDevice, stream);

    surfconv_fused<<<NP_TOT / PTS_WG, THREADS, 0, stream>>>(
        points, coordinate, w1, b1, lin_w, lin_b, out2);

    bn_reduce<<<CC, 256, 0, stream>>>(out2, sums);

    const int total4 = CC * NP_TOT / 4;
    bn_normalize<<<(total4 + 255) / 256, 256, 0, stream>>>(
        out2, sums, gamma, beta, out + XYZ_N);
}